// Matcher_65867618452264
// MI455X (gfx1250) — compile-verified
//
#include <hip/hip_runtime.h>
#include <cstdint>

// Problem constants (match reference)
constexpr int B_   = 64;
constexpr int N_   = 2048;
constexpr int C_   = 558;   // classes
constexpr int CD_  = 559;   // classes + dustbin
constexpr int VPAD = 560;   // padded v row
// log(1/559)
#define LOG_NU (-6.3261499f)

typedef float v2f __attribute__((ext_vector_type(2)));
typedef float v8f __attribute__((ext_vector_type(8)));

// ---------------------------------------------------------------------------
// prep: per-batch visible count -> lm[b]; init log_v = 0
// ---------------------------------------------------------------------------
__global__ void prep_kernel(const int* __restrict__ mask,
                            float* __restrict__ log_v,
                            float* __restrict__ lm_arr) {
    const int b = blockIdx.x;
    const int tid = threadIdx.x;
    int c = 0;
    for (int n = tid; n < N_; n += 256)
        c += (mask[(size_t)b * N_ + n] != 0) ? 1 : 0;
    for (int off = 16; off > 0; off >>= 1)
        c += __shfl_down(c, off, 32);
    __shared__ int wsum[8];
    if ((tid & 31) == 0) wsum[tid >> 5] = c;
    __syncthreads();
    if (tid == 0) {
        int total = 0;
        for (int w = 0; w < 8; ++w) total += wsum[w];
        lm_arr[b] = (total > 0) ? -__logf((float)total) : -__builtin_inff();
    }
    for (int j = tid; j < VPAD; j += 256)
        log_v[(size_t)b * VPAD + j] = 0.0f;
}

// ---------------------------------------------------------------------------
// u-update: log_u[b,n] = log_mu[b,n] - log( sum_j exp(s[b,n,j] + v[b,j]) )
// One wave handles 16 rows. Summation done on the matrix pipe:
// V_WMMA_F32_16X16X4_F32 with A = ones -> D[m,r] = column sums (exact f32).
// log_v staged to LDS via CDNA5 async global->LDS loads.
// ---------------------------------------------------------------------------
__global__ void u_kernel(const float* __restrict__ lp,
                         const int*   __restrict__ mask,
                         const float* __restrict__ lm_arr,
                         const float* __restrict__ log_v,
                         float*       __restrict__ log_u,
                         const float* __restrict__ db_ptr) {
    __shared__ float s_v[VPAD];
    const int b   = blockIdx.y;
    const int tid = threadIdx.x;

    // async-copy v row (560 floats) into LDS (ASYNCcnt path)
    {
        const float* gv = log_v + (size_t)b * VPAD;
        for (int j = tid; j < VPAD; j += 256) {
            unsigned lds_off = (unsigned)(size_t)(&s_v[j]);
            const float* ga = gv + j;
            asm volatile("global_load_async_to_lds_b32 %0, %1, off"
                         :: "v"(lds_off), "v"(ga) : "memory");
        }
        asm volatile("s_wait_asynccnt 0" ::: "memory");
    }
    __syncthreads();

    const float db   = *db_ptr;
    const int   wave = tid >> 5;
    const int   lane = tid & 31;
    const int   half = lane >> 4;     // which 16-wide half of the wave
    const int   r    = lane & 15;     // row-within-tile == output column
    const int   row  = (blockIdx.x * 8 + wave) * 16 + r;

    const float* rowp = lp + ((size_t)b * N_ + row) * (size_t)C_;

    v8f acc0 = {0.f,0.f,0.f,0.f,0.f,0.f,0.f,0.f};
    v8f acc1 = {0.f,0.f,0.f,0.f,0.f,0.f,0.f,0.f};
    v2f ones; ones[0] = 1.0f; ones[1] = 1.0f;

    float e[16];
    // main body: j = 0..543 (17 chunks of 32)
    for (int jc = 0; jc < 544; jc += 32) {
        const int jb = jc + half * 16;
        #pragma unroll
        for (int t = 0; t < 16; t += 2) {
            float2 sv = *reinterpret_cast<const float2*>(rowp + jb + t);
            float2 vv = *reinterpret_cast<const float2*>(&s_v[jb + t]);
            e[t]     = __expf(sv.x + vv.x);
            e[t + 1] = __expf(sv.y + vv.y);
        }
        #pragma unroll
        for (int t = 0; t < 16; t += 4) {
            v2f b0; b0[0] = e[t];     b0[1] = e[t + 1];
            v2f b1; b1[0] = e[t + 2]; b1[1] = e[t + 3];
            acc0 = __builtin_amdgcn_wmma_f32_16x16x4_f32(false, ones, false, b0,
                                                         (short)0, acc0, false, false);
            acc1 = __builtin_amdgcn_wmma_f32_16x16x4_f32(false, ones, false, b1,
                                                         (short)0, acc1, false, false);
        }
    }
    // tail: j = 544..558 (incl. dustbin col 558), rest zero
    {
        const int jb = 544 + half * 16;
        #pragma unroll
        for (int t = 0; t < 16; ++t) {
            const int j = jb + t;
            float val = 0.0f;
            if (j < C_)       val = __expf(rowp[j] + s_v[j]);
            else if (j == C_) val = __expf(db + s_v[j]);
            e[t] = val;
        }
        #pragma unroll
        for (int t = 0; t < 16; t += 4) {
            v2f b0; b0[0] = e[t];     b0[1] = e[t + 1];
            v2f b1; b1[0] = e[t + 2]; b1[1] = e[t + 3];
            acc0 = __builtin_amdgcn_wmma_f32_16x16x4_f32(false, ones, false, b0,
                                                         (short)0, acc0, false, false);
            acc1 = __builtin_amdgcn_wmma_f32_16x16x4_f32(false, ones, false, b1,
                                                         (short)0, acc1, false, false);
        }
    }

    if (half == 0) {
        const float sum = acc0[0] + acc1[0];           // D[0, r] for row `row`
        const size_t idx = (size_t)b * N_ + row;
        const float lmu = (mask[idx] != 0) ? lm_arr[b] : -__builtin_inff();
        log_u[idx] = lmu - __logf(sum);
    }
}

// ---------------------------------------------------------------------------
// v-update: log_v[b,j] = LOG_NU - log( sum_n exp(s[b,n,j] + u[b,n]) )
// Thread-per-column, coalesced over j; u staged in LDS.
// ---------------------------------------------------------------------------
__global__ void v_kernel(const float* __restrict__ lp,
                         const float* __restrict__ log_u,
                         float*       __restrict__ log_v,
                         const float* __restrict__ db_ptr) {
    __shared__ float s_u[N_];
    const int b = blockIdx.y, tid = threadIdx.x;
    const float* up = log_u + (size_t)b * N_;
    for (int n = tid; n < N_; n += 256) s_u[n] = up[n];
    __syncthreads();

    const int j = blockIdx.x * 256 + tid;
    if (j >= CD_) return;

    float s0 = 0.f, s1 = 0.f, s2 = 0.f, s3 = 0.f;
    if (j < C_) {
        const float* colp = lp + (size_t)b * N_ * C_ + j;
        for (int n = 0; n < N_; n += 4) {
            float a0 = colp[(size_t)(n + 0) * C_];
            float a1 = colp[(size_t)(n + 1) * C_];
            float a2 = colp[(size_t)(n + 2) * C_];
            float a3 = colp[(size_t)(n + 3) * C_];
            s0 += __expf(a0 + s_u[n + 0]);
            s1 += __expf(a1 + s_u[n + 1]);
            s2 += __expf(a2 + s_u[n + 2]);
            s3 += __expf(a3 + s_u[n + 3]);
        }
    } else { // dustbin column: score is constant db
        const float db = *db_ptr;
        for (int n = 0; n < N_; n += 4) {
            s0 += __expf(db + s_u[n + 0]);
            s1 += __expf(db + s_u[n + 1]);
            s2 += __expf(db + s_u[n + 2]);
            s3 += __expf(db + s_u[n + 3]);
        }
    }
    const float sum = (s0 + s1) + (s2 + s3);
    log_v[(size_t)b * VPAD + j] = LOG_NU - __logf(sum);
}

// ---------------------------------------------------------------------------
// final: soft = exp(s+u+v), hard = argmax_j soft (first max), dustbin col
// One wave per row; 8 rows per block; v staged in LDS; wave32 shuffle argmax.
// ---------------------------------------------------------------------------
__global__ void final_kernel(const float* __restrict__ lp,
                             const float* __restrict__ log_u,
                             const float* __restrict__ log_v,
                             const float* __restrict__ db_ptr,
                             float* __restrict__ out_soft,
                             float* __restrict__ out_hard,
                             float* __restrict__ out_dust) {
    __shared__ float s_v[VPAD];
    const int tid = threadIdx.x;
    const int b   = blockIdx.x >> 8;          // 256 blocks per batch
    const int n0  = (blockIdx.x & 255) * 8;   // 8 rows per block
    const float* gv = log_v + (size_t)b * VPAD;
    for (int j = tid; j < VPAD; j += 256) s_v[j] = gv[j];
    __syncthreads();

    const int wave = tid >> 5, lane = tid & 31;
    const int n = n0 + wave;
    const size_t rowIdx = (size_t)b * N_ + n;
    const float u  = log_u[rowIdx];
    const float db = *db_ptr;
    const float* rowp = lp + rowIdx * (size_t)C_;
    float* srow = out_soft + rowIdx * (size_t)C_;

    float best = -1.0f;
    int   bidx = 0x7fffffff;
    int j = lane;
    for (; j < C_; j += 32) {
        const float e = __expf(rowp[j] + u + s_v[j]);
        srow[j] = e;
        if (e > best) { best = e; bidx = j; }
    }
    if (j == C_)  // exactly one lane lands on the dustbin column
        out_dust[rowIdx] = __expf(db + u + s_v[C_]);

    for (int off = 16; off > 0; off >>= 1) {
        const float ov = __shfl_down(best, off, 32);
        const int   oi = __shfl_down(bidx, off, 32);
        if (ov > best || (ov == best && oi < bidx)) { best = ov; bidx = oi; }
    }
    if (lane == 0) out_hard[rowIdx] = (float)bidx;
}

// ---------------------------------------------------------------------------
extern "C" void kernel_launch(void* const* d_in, const int* in_sizes, int n_in,
                              void* d_out, int out_size, void* d_ws, size_t ws_size,
                              hipStream_t stream) {
    const float* lp   = (const float*)d_in[0];   // log_probs  (B,N,C) f32
    const int*   mask = (const int*)  d_in[1];   // visible_mask (B,N)
    const float* db   = (const float*)d_in[2];   // dustbin_score scalar

    float* ws    = (float*)d_ws;
    float* log_u = ws;                                  // B*N
    float* log_v = ws + (size_t)B_ * N_;                // B*VPAD
    float* lm    = log_v + (size_t)B_ * VPAD;           // B

    float* out      = (float*)d_out;
    float* out_soft = out;                              // B*N*C
    float* out_hard = out + (size_t)B_ * N_ * C_;       // B*N
    float* out_dust = out_hard + (size_t)B_ * N_;       // B*N

    prep_kernel<<<B_, 256, 0, stream>>>(mask, log_v, lm);

    for (int it = 0; it < 5; ++it) {
        u_kernel<<<dim3(N_ / 128, B_), 256, 0, stream>>>(lp, mask, lm, log_v, log_u, db);
        v_kernel<<<dim3((CD_ + 255) / 256, B_), 256, 0, stream>>>(lp, log_u, log_v, db);
    }

    final_kernel<<<dim3(B_ * N_ / 8), 256, 0, stream>>>(lp, log_u, log_v, db,
                                                        out_soft, out_hard, out_dust);
}